// GATModel_30459908063504
// MI455X (gfx1250) — compile-verified
//
#include <hip/hip_runtime.h>
#include <hip/hip_bf16.h>
#include <math.h>

#define NODES 650
#define NPAD  656      // 41 * 16, so WMMA tiles cover M exactly (EXEC all-ones)
#define DIM   512

typedef __attribute__((ext_vector_type(2))) float v2f;
typedef __attribute__((ext_vector_type(8))) float v8f;

// ---------------- helpers ----------------
__device__ __forceinline__ void atomicMaxF(float* addr, float val) {
  // IEEE-ordering trick: int max for non-negative, uint min for negative
  if (val >= 0.0f) atomicMax((int*)addr, __float_as_int(val));
  else             atomicMin((unsigned int*)addr, __float_as_uint(val));
}

// ---------------- x = concat(x_s, x_t) padded to [NPAD, DIM] ----------------
__global__ void build_x(const float* __restrict__ xs, const float* __restrict__ xt,
                        float* __restrict__ x, int ns, int nt) {
  int i = blockIdx.x * blockDim.x + threadIdx.x;
  if (i >= NPAD * DIM) return;
  int r = i / DIM, c = i % DIM;
  float v = 0.0f;
  if (r < ns)            v = xs[r * DIM + c];
  else if (r < ns + nt)  v = xt[(r - ns) * DIM + c];
  x[i] = v;
}

// ---------------- C[NPAD x DIM] = A[NPAD x DIM] * B[DIM x DIM], f32 WMMA ----
// One wave per 16x16 output tile; 4 waves per block tile along N.
// A 16x4 layout: lanes 0-15 hold rows, VGPR0/1 = K0/K1; lanes 16-31 = K2/K3.
// B 4x16 layout: VGPR0 lanes0-15 = row K0, lanes16-31 = row K1; VGPR1 = K2/K3.
__global__ void gemm_wmma(const float* __restrict__ A, const float* __restrict__ B,
                          float* __restrict__ C) {
  const int lane = threadIdx.x & 31;
  const int wave = threadIdx.x >> 5;
  const int tn   = (blockIdx.x * 4 + wave) * 16;   // N tile (always < DIM)
  const int tm   = blockIdx.y * 16;                // M tile (always < NPAD)
  const int rc   = lane & 15;                      // row for A, col for B/C
  const int hi   = lane >> 4;                      // lane-half select

  v8f acc = {};
  const float* Ar = A + (tm + rc) * DIM;
  const float* Bc = B + tn + rc;
  for (int k = 0; k < DIM; k += 4) {
    v2f a, b;
    a.x = Ar[k + 2 * hi + 0];
    a.y = Ar[k + 2 * hi + 1];
    b.x = Bc[(k + hi) * DIM];
    b.y = Bc[(k + 2 + hi) * DIM];
    acc = __builtin_amdgcn_wmma_f32_16x16x4_f32(false, a, false, b,
                                                (short)0, acc, false, false);
  }
  // C/D layout: VGPR i -> M = tm + hi*8 + i, N = tn + rc
  float* Cp = C + (tm + hi * 8) * DIM + tn + rc;
#pragma unroll
  for (int i = 0; i < 8; i++) Cp[i * DIM] = acc[i];
}

// ---------------- per-node attention logits: al_s = h.a_src, al_d = h.a_dst -
__global__ void node_logits(const float* __restrict__ h, const float* __restrict__ asrc,
                            const float* __restrict__ adst,
                            float* __restrict__ als, float* __restrict__ ald) {
  int lane = threadIdx.x & 31;
  int wave = threadIdx.x >> 5;
  int n = blockIdx.x * 8 + wave;
  if (n >= NODES) return;
  const float* hr = h + n * DIM;
  float s = 0.0f, d = 0.0f;
  for (int c = lane; c < DIM; c += 32) {
    float v = hr[c];
    s += v * asrc[c];
    d += v * adst[c];
  }
  for (int off = 16; off > 0; off >>= 1) {
    s += __shfl_xor(s, off, 32);
    d += __shfl_xor(d, off, 32);
  }
  if (lane == 0) { als[n] = s; ald[n] = d; }
}

// ---------------- segment state init ----------------
__global__ void init_seg(float* __restrict__ segmax, float* __restrict__ segsum) {
  int i = blockIdx.x * blockDim.x + threadIdx.x;
  if (i < NODES) { segmax[i] = -3.0e38f; segsum[i] = 0.0f; }
}

// agg initialized with bias (segment_sum + b fused); pad rows -> 0
__global__ void init_agg(float* __restrict__ agg, const float* __restrict__ b) {
  int i = blockIdx.x * blockDim.x + threadIdx.x;
  if (i >= NPAD * DIM) return;
  int r = i / DIM, c = i % DIM;
  agg[i] = (r < NODES) ? b[c] : 0.0f;
}

// ---------------- edge pass 1: segment max of leaky_relu(logits) ------------
__global__ void edge_max(const int* __restrict__ ei, int E,
                         const float* __restrict__ als, const float* __restrict__ ald,
                         float* __restrict__ segmax) {
  int e = blockIdx.x * blockDim.x + threadIdx.x;
  int ET = E + NODES;
  if (e >= ET) return;
  int s, d;
  if (e < E) { s = ei[e]; d = ei[E + e]; } else { s = d = e - E; }
  float v = als[s] + ald[d];
  v = (v > 0.0f) ? v : 0.2f * v;
  atomicMaxF(&segmax[d], v);
}

// ---------------- edge pass 2: ex = exp(e - max[dst]); segment sum ----------
__global__ void edge_expsum(const int* __restrict__ ei, int E,
                            const float* __restrict__ als, const float* __restrict__ ald,
                            const float* __restrict__ segmax,
                            float* __restrict__ exbuf, float* __restrict__ segsum) {
  int e = blockIdx.x * blockDim.x + threadIdx.x;
  int ET = E + NODES;
  if (e >= ET) return;
  int s, d;
  if (e < E) { s = ei[e]; d = ei[E + e]; } else { s = d = e - E; }
  float v = als[s] + ald[d];
  v = (v > 0.0f) ? v : 0.2f * v;
  float ex = __expf(v - segmax[d]);
  exbuf[e] = ex;
  atomicAdd(&segsum[d], ex);
}

// ---------------- edge pass 3: agg[dst] += alpha * h[src] (wave per edge) ---
__global__ void edge_agg(const int* __restrict__ ei, int E,
                         const float* __restrict__ exbuf, const float* __restrict__ segsum,
                         const float* __restrict__ h, float* __restrict__ agg) {
  int lane = threadIdx.x & 31;
  int wave = threadIdx.x >> 5;
  int e = blockIdx.x * 8 + wave;
  int ET = E + NODES;
  if (e >= ET) return;
  int s, d;
  if (e < E) { s = ei[e]; d = ei[E + e]; } else { s = d = e - E; }
  float alpha = exbuf[e] / (segsum[d] + 1e-16f);
  const float* hs = h + s * DIM;
  float* ad = agg + d * DIM;
  for (int c = lane; c < DIM; c += 32) atomicAdd(&ad[c], alpha * hs[c]);
}

// ---------------- elementwise (leaky) relu ----------------
__global__ void post_act(float* __restrict__ a, float slope) {
  int i = blockIdx.x * blockDim.x + threadIdx.x;
  if (i >= NPAD * DIM) return;
  float v = a[i];
  a[i] = (v > 0.0f) ? v : slope * v;
}

// ---------------- final FC (flat view [512,650]) + sigmoid ------------------
__global__ void fc_sigmoid(const float* __restrict__ y, const float* __restrict__ w,
                           const float* __restrict__ bias, float* __restrict__ out) {
  int lane = threadIdx.x & 31;
  int wave = threadIdx.x >> 5;
  int j = blockIdx.x * 8 + wave;
  if (j >= 512) return;
  const float* yr = y + j * NODES;   // torch .view: row j is flat[j*650 .. )
  float s = 0.0f;
  for (int k = lane; k < NODES; k += 32) s += yr[k] * w[k];
  for (int off = 16; off > 0; off >>= 1) s += __shfl_xor(s, off, 32);
  if (lane == 0) out[j] = 1.0f / (1.0f + __expf(-(s + bias[0])));
}

// ============================================================================
extern "C" void kernel_launch(void* const* d_in, const int* in_sizes, int n_in,
                              void* d_out, int out_size, void* d_ws, size_t ws_size,
                              hipStream_t stream) {
  const float* xs    = (const float*)d_in[0];
  const float* xt    = (const float*)d_in[1];
  const int*   ei    = (const int*)d_in[2];
  // d_in[3] distances, d_in[4] batch: unused by the reference
  const float* W1    = (const float*)d_in[5];
  const float* asrc1 = (const float*)d_in[6];
  const float* adst1 = (const float*)d_in[7];
  const float* b1    = (const float*)d_in[8];
  const float* W4    = (const float*)d_in[9];
  const float* asrc4 = (const float*)d_in[10];
  const float* adst4 = (const float*)d_in[11];
  const float* b4    = (const float*)d_in[12];
  const float* fcw   = (const float*)d_in[13];
  const float* fcb   = (const float*)d_in[14];
  float* out = (float*)d_out;

  const int ns = in_sizes[0] / DIM;       // 400
  const int nt = in_sizes[1] / DIM;       // 250
  const int E  = in_sizes[2] / 2;         // 150000
  const int ET = E + NODES;               // + self loops

  // workspace layout (floats)
  float* ws   = (float*)d_ws;
  float* xA   = ws;                       // [NPAD, DIM] input / layer-2 agg
  float* hB   = xA + NPAD * DIM;          // [NPAD, DIM] GEMM output (both layers)
  float* aggC = hB + NPAD * DIM;          // [NPAD, DIM] layer-1 agg
  float* als  = aggC + NPAD * DIM;        // [NPAD]
  float* ald  = als + NPAD;
  float* smax = ald + NPAD;
  float* ssum = smax + NPAD;
  float* exb  = ssum + NPAD;              // [ET]

  const int ELEMS = NPAD * DIM;
  dim3 bElem(256), gElem((ELEMS + 255) / 256);
  dim3 bGemm(128), gGemm(DIM / 64, NPAD / 16);
  dim3 bNode(256), gNode((NODES + 7) / 8);
  dim3 bEdge(256), gEdge((ET + 255) / 256);
  dim3 gEdgeW((ET + 7) / 8);              // wave-per-edge
  dim3 gSeg((NODES + 255) / 256);

  // ---- build input ----
  build_x<<<gElem, bElem, 0, stream>>>(xs, xt, xA, ns, nt);

  // ---- layer 1 ----
  gemm_wmma<<<gGemm, bGemm, 0, stream>>>(xA, W1, hB);
  node_logits<<<gNode, bNode, 0, stream>>>(hB, asrc1, adst1, als, ald);
  init_seg<<<gSeg, bElem, 0, stream>>>(smax, ssum);
  init_agg<<<gElem, bElem, 0, stream>>>(aggC, b1);
  edge_max<<<gEdge, bEdge, 0, stream>>>(ei, E, als, ald, smax);
  edge_expsum<<<gEdge, bEdge, 0, stream>>>(ei, E, als, ald, smax, exb, ssum);
  edge_agg<<<gEdgeW, bEdge, 0, stream>>>(ei, E, exb, ssum, hB, aggC);
  post_act<<<gElem, bElem, 0, stream>>>(aggC, 0.0f);          // relu

  // ---- layer 2 ----
  gemm_wmma<<<gGemm, bGemm, 0, stream>>>(aggC, W4, hB);
  node_logits<<<gNode, bNode, 0, stream>>>(hB, asrc4, adst4, als, ald);
  init_seg<<<gSeg, bElem, 0, stream>>>(smax, ssum);
  init_agg<<<gElem, bElem, 0, stream>>>(xA, b4);              // reuse xA as agg2
  edge_max<<<gEdge, bEdge, 0, stream>>>(ei, E, als, ald, smax);
  edge_expsum<<<gEdge, bEdge, 0, stream>>>(ei, E, als, ald, smax, exb, ssum);
  edge_agg<<<gEdgeW, bEdge, 0, stream>>>(ei, E, exb, ssum, hB, xA);
  post_act<<<gElem, bElem, 0, stream>>>(xA, 0.01f);           // leaky_relu

  // ---- final fc + sigmoid ----
  fc_sigmoid<<<64, 256, 0, stream>>>(xA, fcw, fcb, out);
}